// RainReservoirEncoder_59150289601184
// MI455X (gfx1250) — compile-verified
//
#include <hip/hip_runtime.h>

// RainReservoirEncoder on gfx1250 (MI455X, wave32):
//   per 16-batch tile: X' = tanh(W * X + w_in * u^T), T=4096 serial steps.
//   W  -> WMMA A operand (4 constant 16x4 f32 fragments, preloaded once)
//   X  -> WMMA B operand (hidden rows x 16 batch cols), rebuilt each step
//   input term -> C operand of chain 1; chain 2 seeded with inline-0 C.
// Per tile-step: two independent 2-deep v_wmma_f32_16x16x4_f32 chains merged
// with one v8f add (4x v_pk_add_f32), hardware v_tanh_f32, then D->B relayout
// via v_permlane16_swap_b32 (1 instr produces BOTH fragments of a row pair;
// permlanex16+cndmask fallback if the swap builtin is unavailable).
// Two independent batch tiles per wave, interleaved for in-order issue ILP.

#define HIDDEN   16
#define T_LEN    4096
#define B_TOT    8192
#define TILES_PER_WAVE 2

typedef __attribute__((ext_vector_type(2))) float    v2f;
typedef __attribute__((ext_vector_type(4))) float    v4f;
typedef __attribute__((ext_vector_type(8))) float    v8f;

__device__ __forceinline__ float htanh(float x) {
#if __has_builtin(__builtin_amdgcn_tanhf)
  return __builtin_amdgcn_tanhf(x);
#else
  float y;
  asm("v_tanh_f32 %0, %1" : "=v"(y) : "v"(x));
  return y;
#endif
}

#if __has_builtin(__builtin_amdgcn_permlane16_swap)
#define HAVE_PERMLANE_SWAP 1
#else
#define HAVE_PERMLANE_SWAP 0
// lane i <-> lane i^16 (identity lane-selects)
__device__ __forceinline__ float halfswap(float x) {
  int xi = __float_as_int(x);
  int r = __builtin_amdgcn_permlanex16(xi, xi, 0x76543210, 0xfedcba98, false, false);
  return __int_as_float(r);
}
#endif

struct Tile {
  // B fragments: bf[2k+i] = VGPR i of K-chunk k = rows (4k+i | 4k+2+i)
  float bf[8];
};

// One recurrence step for TWO independent tiles, interleaved for ILP.
__device__ __forceinline__ void step2(Tile& a, Tile& b, float ua, float ub,
                                      const v2f* __restrict__ wfrag,
                                      const float* __restrict__ winj, bool hi) {
  // Chain-1 C = input term: C[h',batch] = w_in[h'] * u[batch]
  v8f ca1, cb1;
#pragma unroll
  for (int j = 0; j < 8; ++j) { ca1[j] = ua * winj[j]; cb1[j] = ub * winj[j]; }
  v8f ca2 = {}, cb2 = {};  // chain-2 C: inline 0

  // Four independent 2-deep WMMA chains (2 per tile), issue-interleaved.
#pragma unroll
  for (int k = 0; k < 2; ++k) {
    v2f ba1; ba1.x = a.bf[2 * k + 0]; ba1.y = a.bf[2 * k + 1];
    v2f ba2; ba2.x = a.bf[2 * k + 4]; ba2.y = a.bf[2 * k + 5];
    v2f bb1; bb1.x = b.bf[2 * k + 0]; bb1.y = b.bf[2 * k + 1];
    v2f bb2; bb2.x = b.bf[2 * k + 4]; bb2.y = b.bf[2 * k + 5];
    ca1 = __builtin_amdgcn_wmma_f32_16x16x4_f32(false, wfrag[k + 0], false, ba1,
                                                (short)0, ca1, false, false);
    ca2 = __builtin_amdgcn_wmma_f32_16x16x4_f32(false, wfrag[k + 2], false, ba2,
                                                (short)0, ca2, false, false);
    cb1 = __builtin_amdgcn_wmma_f32_16x16x4_f32(false, wfrag[k + 0], false, bb1,
                                                (short)0, cb1, false, false);
    cb2 = __builtin_amdgcn_wmma_f32_16x16x4_f32(false, wfrag[k + 2], false, bb2,
                                                (short)0, cb2, false, false);
  }

  // Merge chains (v_pk_add_f32 x4 each) and apply hardware tanh.
  float sa[8], sb[8];
  v8f accA = ca1 + ca2;
  v8f accB = cb1 + cb2;
#pragma unroll
  for (int j = 0; j < 8; ++j) { sa[j] = htanh(accA[j]); sb[j] = htanh(accB[j]); }

  // D layout: s[j] = rows (j | j+8). Need bf pairs (r|r+2) and (r+8|r+10).
#if HAVE_PERMLANE_SWAP
  // v_permlane16_swap_b32(vdst=s[r], vsrc=s[r+2]):
  //   new vdst = rows(r | r+2), new vsrc = rows(r+8 | r+10) -- both fragments
  //   of the pair in ONE instruction, in place.
#pragma unroll
  for (int p = 0; p < 4; ++p) {
    const int r  = (p & 1) + 4 * (p >> 1);      // 0,1,4,5
    const int lo = (p & 1) + 2 * (p >> 1);      // bf index of rows(r|r+2): 0,1,2,3
    {
      auto ra = __builtin_amdgcn_permlane16_swap(__float_as_uint(sa[r]),
                                                 __float_as_uint(sa[r + 2]),
                                                 false, false);
      a.bf[lo] = __uint_as_float(ra[0]);        // rows r   | r+2
      a.bf[lo + 4] = __uint_as_float(ra[1]);    // rows r+8 | r+10
    }
    {
      auto rb = __builtin_amdgcn_permlane16_swap(__float_as_uint(sb[r]),
                                                 __float_as_uint(sb[r + 2]),
                                                 false, false);
      b.bf[lo] = __uint_as_float(rb[0]);
      b.bf[lo + 4] = __uint_as_float(rb[1]);
    }
  }
#else
#pragma unroll
  for (int p = 0; p < 4; ++p) {
    const int r  = (p & 1) + 4 * (p >> 1);
    const int lo = (p & 1) + 2 * (p >> 1);
    float haL = halfswap(sa[r + 2]), haH = halfswap(sa[r]);
    float hbL = halfswap(sb[r + 2]), hbH = halfswap(sb[r]);
    a.bf[lo]     = hi ? haL : sa[r];      // rows r   | r+2
    a.bf[lo + 4] = hi ? sa[r + 2] : haH;  // rows r+8 | r+10
    b.bf[lo]     = hi ? hbL : sb[r];
    b.bf[lo + 4] = hi ? sb[r + 2] : hbH;
  }
#endif
}

__global__ __launch_bounds__(128) void rain_reservoir_kernel(
    const float* __restrict__ rain,   // (B, T, 1)
    const float* __restrict__ w_in,   // (16, 1)
    const float* __restrict__ w,      // (16, 16) row-major W[h'][h]
    float* __restrict__ out) {        // (B, 16)
  const int lane = threadIdx.x & 31;
  const int col  = lane & 15;        // batch column within tile
  const int half = lane >> 4;        // 0: lanes 0-15, 1: lanes 16-31
  const bool hi  = (half != 0);

  const int wave  = (blockIdx.x * blockDim.x + threadIdx.x) >> 5;
  const int tile0 = wave * TILES_PER_WAVE;
  const int b0    = (tile0 + 0) * 16;
  const int b1    = (tile0 + 1) * 16;

  // Constant W fragments (A operand, 16x4 f32 per K-chunk):
  //   VGPR0: K = 4k + 2*half ; VGPR1: K = 4k + 1 + 2*half ; M = lane&15
  v2f wfrag[4];
#pragma unroll
  for (int k = 0; k < 4; ++k) {
    wfrag[k].x = w[col * HIDDEN + 4 * k + 2 * half];
    wfrag[k].y = w[col * HIDDEN + 4 * k + 1 + 2 * half];
  }
  // w_in broadcast matching C/D row layout: winj[j] = w_in[j + 8*half]
  float winj[8];
#pragma unroll
  for (int j = 0; j < 8; ++j) winj[j] = w_in[j + 8 * half];

  Tile t0, t1;
#pragma unroll
  for (int j = 0; j < 8; ++j) { t0.bf[j] = 0.0f; t1.bf[j] = 0.0f; }

  // Per-lane input streams; both half-waves read the same batch row so u is
  // lane-duplicated exactly as the B/C layouts require. Non-temporal: the
  // 128 MiB input is touched exactly once.
  const v4f* u0p = (const v4f*)(rain + (size_t)(b0 + col) * T_LEN);
  const v4f* u1p = (const v4f*)(rain + (size_t)(b1 + col) * T_LEN);

  v4f u0 = __builtin_nontemporal_load(u0p);
  v4f u1 = __builtin_nontemporal_load(u1p);
  ++u0p; ++u1p;

  // Main loop: prefetch next 4 inputs per tile, run 4 interleaved steps.
  for (int t = 0; t < T_LEN - 4; t += 4) {
    v4f n0 = __builtin_nontemporal_load(u0p);
    v4f n1 = __builtin_nontemporal_load(u1p);
    ++u0p; ++u1p;

    step2(t0, t1, u0.x, u1.x, wfrag, winj, hi);
    step2(t0, t1, u0.y, u1.y, wfrag, winj, hi);
    step2(t0, t1, u0.z, u1.z, wfrag, winj, hi);
    step2(t0, t1, u0.w, u1.w, wfrag, winj, hi);

    u0 = n0;
    u1 = n1;
  }
  // Final 4 steps (no prefetch).
  step2(t0, t1, u0.x, u1.x, wfrag, winj, hi);
  step2(t0, t1, u0.y, u1.y, wfrag, winj, hi);
  step2(t0, t1, u0.z, u1.z, wfrag, winj, hi);
  step2(t0, t1, u0.w, u1.w, wfrag, winj, hi);

  // Final state straight from the B fragments:
  // bf[j] holds rows (4*(j>>1) + (j&1) | +2) -> row = 4*(j>>1)+(j&1)+2*half.
#pragma unroll
  for (int j = 0; j < 8; ++j) {
    const int row = 4 * (j >> 1) + (j & 1) + 2 * half;
    out[(size_t)(b0 + col) * HIDDEN + row] = t0.bf[j];
    out[(size_t)(b1 + col) * HIDDEN + row] = t1.bf[j];
  }
}

extern "C" void kernel_launch(void* const* d_in, const int* in_sizes, int n_in,
                              void* d_out, int out_size, void* d_ws, size_t ws_size,
                              hipStream_t stream) {
  const float* rain = (const float*)d_in[0];  // (8192, 4096, 1) f32
  const float* w_in = (const float*)d_in[1];  // (16, 1) f32
  const float* w    = (const float*)d_in[2];  // (16, 16) f32
  float* out        = (float*)d_out;          // (8192, 16) f32

  const int tiles   = B_TOT / 16;                 // 512
  const int waves   = tiles / TILES_PER_WAVE;     // 256
  const int threads = 128;                        // 4 waves / block
  const int blocks  = (waves * 32) / threads;     // 64

  rain_reservoir_kernel<<<blocks, threads, 0, stream>>>(rain, w_in, w, out);
}